// MoEGate_24180665876612
// MI455X (gfx1250) — compile-verified
//
#include <hip/hip_runtime.h>

// MoE gate for MI455X (gfx1250): fp32 WMMA (V_WMMA_F32_16X16X4_F32) logits,
// streaming HBM-bound design (~128 MB read, ~5.5us floor @ 23.3 TB/s).

typedef float v2f __attribute__((ext_vector_type(2)));
typedef float v8f __attribute__((ext_vector_type(8)));

#define EXPERTS 4
#define TOPK 2
#define HDIM 1024
#define WAVES_PER_BLOCK 8
#define BLOCK_THREADS (WAVES_PER_BLOCK * 32)
#define TILE_TOKENS 16
#define TOKENS_PER_BLOCK (WAVES_PER_BLOCK * TILE_TOKENS)

__global__ __launch_bounds__(BLOCK_THREADS)
void moe_gate_kernel(const float* __restrict__ x,      // [n_tokens, HDIM]
                     const float* __restrict__ w,      // [EXPERTS, HDIM]
                     float* __restrict__ out_idx,      // [n_tokens, TOPK] (as float)
                     float* __restrict__ out_wgt,      // [n_tokens, TOPK]
                     float* __restrict__ partials)     // [n_tiles] score-sum partials
{
    __shared__ float lds_logits[WAVES_PER_BLOCK][TILE_TOKENS][EXPERTS + 1];

    const int lane = threadIdx.x & 31;
    const int wave = threadIdx.x >> 5;
    const int tile = blockIdx.x * WAVES_PER_BLOCK + wave;
    const int tok0 = tile * TILE_TOKENS;

    // CDNA5 fp32 16x4 A-layout: lanes 0-15 hold rows M=0..15 at K={0,1};
    // lanes 16-31 hold the same rows at K={2,3}. B mirrors with N<->M.
    const int row   = lane & 15;          // token-in-tile (A) / output column (B)
    const int khalf = (lane >> 4) << 1;   // 0 or 2

    const float* aptr = x + (size_t)(tok0 + row) * HDIM + khalf;
    // Columns 4..15 alias experts 0..3 (in-bounds loads); their outputs are ignored.
    const float* bptr = w + (size_t)(row & (EXPERTS - 1)) * HDIM + khalf;

    v8f c = {0.f, 0.f, 0.f, 0.f, 0.f, 0.f, 0.f, 0.f};

    for (int kk = 0; kk < HDIM; kk += 32) {
        __builtin_prefetch(aptr + kk + 512, 0, 0);   // global_prefetch_b8, ~2KB ahead
        #pragma unroll
        for (int q = 0; q < 32; q += 4) {
            v2f a = *(const v2f*)(aptr + kk + q);
            v2f b = *(const v2f*)(bptr + kk + q);
            // (neg_a, A, neg_b, B, c_mod, C, reuse_a, reuse_b)
            c = __builtin_amdgcn_wmma_f32_16x16x4_f32(
                    false, a, false, b, (short)0, c, false, false);
        }
    }

    // C/D 16x16 f32 layout: VGPR j holds row M = j + (lane<16 ? 0 : 8), col N = lane%16.
    // Scatter the 4 valid expert columns to LDS so one lane owns one token.
    if (row < EXPERTS) {
        const int mbase = (lane >> 4) * 8;
        #pragma unroll
        for (int j = 0; j < 8; ++j)
            lds_logits[wave][mbase + j][row] = c[j];
    }
    __syncthreads();

    float psum = 0.f;
    if (lane < TILE_TOKENS) {
        const float l0 = lds_logits[wave][lane][0];
        const float l1 = lds_logits[wave][lane][1];
        const float l2 = lds_logits[wave][lane][2];
        const float l3 = lds_logits[wave][lane][3];

        const float m  = fmaxf(fmaxf(l0, l1), fmaxf(l2, l3));
        float p[EXPERTS];
        p[0] = expf(l0 - m);
        p[1] = expf(l1 - m);
        p[2] = expf(l2 - m);
        p[3] = expf(l3 - m);
        const float Z   = p[0] + p[1] + p[2] + p[3];
        const float inv = 1.f / Z;
        #pragma unroll
        for (int e = 0; e < EXPERTS; ++e) p[e] *= inv;

        // top-1 (strict > keeps lowest index on ties, matching lax.top_k)
        int   i0 = 0;
        float v0 = p[0];
        #pragma unroll
        for (int e = 1; e < EXPERTS; ++e)
            if (p[e] > v0) { v0 = p[e]; i0 = e; }
        // top-2 excluding i0
        int   i1 = -1;
        float v1 = -1.f;   // probs are >= 0
        #pragma unroll
        for (int e = 0; e < EXPERTS; ++e)
            if (e != i0 && p[e] > v1) { v1 = p[e]; i1 = e; }

        const size_t g = (size_t)(tok0 + lane) * TOPK;
        out_idx[g + 0] = (float)i0;
        out_idx[g + 1] = (float)i1;
        out_wgt[g + 0] = v0;
        out_wgt[g + 1] = v1;

        psum = p[0] + p[1] + p[2] + p[3];   // token's softmax sum (for pi)
    }

    // wave32 reduction of psum (lanes >= 16 contribute 0)
    #pragma unroll
    for (int off = 16; off > 0; off >>= 1)
        psum += __shfl_down(psum, off, 32);
    if (lane == 0)
        partials[tile] = psum;
}

__global__ __launch_bounds__(256)
void moe_gate_finalize(const float* __restrict__ partials, int n_partials,
                       int n_tokens, float* __restrict__ aux_out)
{
    __shared__ float red[256];
    float s = 0.f;
    for (int i = threadIdx.x; i < n_partials; i += 256)
        s += partials[i];
    red[threadIdx.x] = s;
    __syncthreads();
    #pragma unroll
    for (int stride = 128; stride > 0; stride >>= 1) {
        if (threadIdx.x < stride) red[threadIdx.x] += red[threadIdx.x + stride];
        __syncthreads();
    }
    if (threadIdx.x == 0) {
        // aux = ALPHA * pi * sum(fi); sum(fi) = E * sum(ce) = E exactly,
        // pi = (sum of all scores) / (N*E)  =>  aux = ALPHA * total / N.
        aux_out[0] = 0.01f * (red[0] / (float)n_tokens);
    }
}

extern "C" void kernel_launch(void* const* d_in, const int* in_sizes, int n_in,
                              void* d_out, int out_size, void* d_ws, size_t ws_size,
                              hipStream_t stream) {
    const float* x = (const float*)d_in[0];   // hidden_states [8,4096,1024] fp32
    const float* w = (const float*)d_in[1];   // weight [4,1024] fp32

    const int n_tokens = in_sizes[0] / HDIM;  // 32768
    const int n_tiles  = n_tokens / TILE_TOKENS;
    const int blocks   = n_tokens / TOKENS_PER_BLOCK;

    float* out      = (float*)d_out;
    float* out_idx  = out;                                   // [N, 2] as float
    float* out_wgt  = out + (size_t)n_tokens * TOPK;         // [N, 2]
    float* aux      = out + (size_t)n_tokens * TOPK * 2;     // scalar
    float* partials = (float*)d_ws;                          // [n_tiles]

    moe_gate_kernel<<<blocks, BLOCK_THREADS, 0, stream>>>(x, w, out_idx, out_wgt, partials);
    moe_gate_finalize<<<1, 256, 0, stream>>>(partials, n_tiles, n_tokens, aux);
}